// AdaptiveDownsampling_60052232733253
// MI455X (gfx1250) — compile-verified
//
#include <hip/hip_runtime.h>
#include <hip/hip_bf16.h>
#include <stdint.h>

// ---------------------------------------------------------------------------
// AdaptiveDownsampling: farthest point sampling (ratio=0.5) + gather.
//   in0: points   [B=8, N=8192, 3]  f32
//   in1: features [B=8, N=8192, C=256] f32
//   out: dp [B, M=4096, 3] ++ df [B, M, 256]   (flat concat, f32)
// Strategy (MI455X / gfx1250):
//   * 1 workgroup (1024 thr = 32 wave32 waves) per batch; the batch's 96KB of
//     points live in LDS (320KB/WGP) for the whole sequential FPS loop.
//   * Points staged global->LDS with GLOBAL_LOAD_ASYNC_TO_LDS_B128 (ASYNCcnt),
//     completion via s_wait_asynccnt.
//   * Per-iteration argmax: register-level max over 8 owned points, wave32
//     shfl_xor tree, then cross-wave LDS reduce. Tie-break = smaller index
//     (matches jnp.argmax first-occurrence).
//   * Feature gather is a separate fully parallel float4 copy kernel.
// ---------------------------------------------------------------------------

#define B_   8
#define N_   8192
#define C_   256
#define M_   4096
#define T_   1024
#define K_   (N_ / T_)   // 8 points per thread

#if defined(__AMDGCN__) && __has_builtin(__builtin_amdgcn_global_load_async_to_lds_b128) && __has_builtin(__builtin_amdgcn_s_wait_asynccnt)
#define HAVE_ASYNC_LDS 1
#else
#define HAVE_ASYNC_LDS 0
#endif

typedef int i32x4 __attribute__((ext_vector_type(4)));

__global__ __launch_bounds__(T_) void fps_kernel(const float* __restrict__ pts,
                                                 int* __restrict__ idx_out) {
  __shared__ float s_pts[N_ * 3];   // 96 KB: whole batch's points
  __shared__ float s_wv[32];        // per-wave partial max value
  __shared__ int   s_wi[32];        // per-wave partial argmax index
  __shared__ float s_last[3];       // coords of most recently selected point

  const int b = blockIdx.x;
  const int t = threadIdx.x;
  const float* bp = pts + (size_t)b * N_ * 3;

  // ---- Stage points into LDS --------------------------------------------
#if HAVE_ASYNC_LDS
  {
    // Each thread copies 24 floats (96 B) = 6 x b128, 16B-aligned both sides.
    __attribute__((address_space(1))) i32x4* g =
        (__attribute__((address_space(1))) i32x4*)(bp + t * 24);
    __attribute__((address_space(3))) i32x4* l =
        (__attribute__((address_space(3))) i32x4*)(&s_pts[t * 24]);
#pragma unroll
    for (int i = 0; i < 6; ++i) {
      __builtin_amdgcn_global_load_async_to_lds_b128(g + i, l + i, 0, 0);
    }
    __builtin_amdgcn_s_wait_asynccnt(0);
  }
#else
  for (int i = t; i < N_ * 3; i += T_) s_pts[i] = bp[i];
#endif
  __syncthreads();

  // ---- Pull owned points into registers (DS reads) ----------------------
  const int base = t * K_;
  float px[K_], py[K_], pz[K_], mind[K_];
#pragma unroll
  for (int k = 0; k < K_; ++k) {
    px[k] = s_pts[(base + k) * 3 + 0];
    py[k] = s_pts[(base + k) * 3 + 1];
    pz[k] = s_pts[(base + k) * 3 + 2];
    mind[k] = 1e10f;
  }

  float lx = s_pts[0], ly = s_pts[1], lz = s_pts[2];   // first pick = index 0
  if (t == 0) idx_out[(size_t)b * M_] = 0;

  const int wave = t >> 5;
  const int lane = t & 31;

  for (int it = 1; it < M_; ++it) {
    // Update running min distance; track local argmax (first-max within thread).
    float bestv = -1.0f;
    int   besti = base;
#pragma unroll
    for (int k = 0; k < K_; ++k) {
      float dx = px[k] - lx;
      float dy = py[k] - ly;
      float dz = pz[k] - lz;
      float d = dx * dx + dy * dy + dz * dz;
      float m = fminf(mind[k], d);
      mind[k] = m;
      if (m > bestv) { bestv = m; besti = base + k; }
    }
    // wave32 shuffle reduction (prefer larger value; tie -> smaller index)
#pragma unroll
    for (int off = 16; off > 0; off >>= 1) {
      float ov = __shfl_xor(bestv, off, 32);
      int   oi = __shfl_xor(besti, off, 32);
      if (ov > bestv || (ov == bestv && oi < besti)) { bestv = ov; besti = oi; }
    }
    if (lane == 0) { s_wv[wave] = bestv; s_wi[wave] = besti; }
    __syncthreads();

    if (wave == 0) {
      float v = s_wv[lane];
      int   i2 = s_wi[lane];
#pragma unroll
      for (int off = 16; off > 0; off >>= 1) {
        float ov = __shfl_xor(v, off, 32);
        int   oi = __shfl_xor(i2, off, 32);
        if (ov > v || (ov == v && oi < i2)) { v = ov; i2 = oi; }
      }
      if (lane == 0) {
        s_last[0] = s_pts[i2 * 3 + 0];
        s_last[1] = s_pts[i2 * 3 + 1];
        s_last[2] = s_pts[i2 * 3 + 2];
        idx_out[(size_t)b * M_ + it] = i2;
      }
    }
    __syncthreads();
    lx = s_last[0];
    ly = s_last[1];
    lz = s_last[2];
  }
}

// Gather selected point coords: dp[b,m,3] = points[b, idx[b,m], :]
__global__ __launch_bounds__(256) void gather_pts_kernel(const float* __restrict__ pts,
                                                         const int* __restrict__ idx,
                                                         float* __restrict__ dp) {
  int i = blockIdx.x * blockDim.x + threadIdx.x;   // over B*M rows
  if (i >= B_ * M_) return;
  int b = i / M_;
  int j = idx[i];
  const float* src = pts + ((size_t)b * N_ + j) * 3;
  float* dst = dp + (size_t)i * 3;
  dst[0] = src[0];
  dst[1] = src[1];
  dst[2] = src[2];
}

// Gather selected feature rows: df[b,m,:] = features[b, idx[b,m], :]
// 256 thr/block = 4 rows/block, 64 lanes x float4 per 1KB row (coalesced b128).
__global__ __launch_bounds__(256) void gather_feat_kernel(const float* __restrict__ feat,
                                                          const int* __restrict__ idx,
                                                          float* __restrict__ df) {
  int row = blockIdx.x * 4 + (threadIdx.x >> 6);    // global row in [0, B*M)
  int c4  = threadIdx.x & 63;
  int b   = row >> 12;                               // row / 4096
  int j   = idx[row];
  const float4* src = (const float4*)(feat + ((size_t)b * N_ + j) * C_);
  float4* dst = (float4*)(df + (size_t)row * C_);
  dst[c4] = src[c4];
}

extern "C" void kernel_launch(void* const* d_in, const int* in_sizes, int n_in,
                              void* d_out, int out_size, void* d_ws, size_t ws_size,
                              hipStream_t stream) {
  (void)in_sizes; (void)n_in; (void)out_size; (void)ws_size;
  const float* points   = (const float*)d_in[0];
  const float* features = (const float*)d_in[1];
  int*   idx = (int*)d_ws;                 // [B, M] selected indices
  float* dp  = (float*)d_out;              // [B, M, 3]
  float* df  = dp + (size_t)B_ * M_ * 3;   // [B, M, C]

  fps_kernel<<<B_, T_, 0, stream>>>(points, idx);
  gather_pts_kernel<<<(B_ * M_ + 255) / 256, 256, 0, stream>>>(points, idx, dp);
  gather_feat_kernel<<<(B_ * M_) / 4, 256, 0, stream>>>(features, idx, df);
}